// NodeEncoder_90701119357583
// MI455X (gfx1250) — compile-verified
//
#include <hip/hip_runtime.h>
#include <math.h>

// ---------------------------------------------------------------------------
// Graph Transformer (3x TransformerConv + LN + graph mean-pool) for gfx1250.
// Dense projections use V_WMMA_F32_16X16X4_F32 (fp32 accuracy), fully
// unrolled (in_c templated) with 2 output tiles per wave for A-fragment reuse.
// Edge phase uses float4 gathers + hardware f32 atomics on L2-resident rows.
// ---------------------------------------------------------------------------

typedef float v2f __attribute__((ext_vector_type(2)));
typedef float v8f __attribute__((ext_vector_type(8)));

#define AADD(p, v)  __hip_atomic_fetch_add((p), (v), __ATOMIC_RELAXED, __HIP_MEMORY_SCOPE_AGENT)
#define AMAXU(p, v) __hip_atomic_fetch_max((p), (v), __ATOMIC_RELAXED, __HIP_MEMORY_SCOPE_AGENT)

// Monotonic order-preserving float->uint encoding for atomic max.
__device__ __forceinline__ unsigned encF(float f) {
  unsigned u = __float_as_uint(f);
  return (u & 0x80000000u) ? ~u : (u | 0x80000000u);
}
__device__ __forceinline__ float decF(unsigned k) {
  return (k & 0x80000000u) ? __uint_as_float(k ^ 0x80000000u)
                           : __uint_as_float(~k);
}
#define ENC_NEG_INF 0x007FFFFFu  // encF(-inf)

__device__ __forceinline__ float waveReduceAdd(float p) {
  p += __shfl_xor(p, 1, 32);
  p += __shfl_xor(p, 2, 32);
  p += __shfl_xor(p, 4, 32);
  p += __shfl_xor(p, 8, 32);
  p += __shfl_xor(p, 16, 32);
  return p;
}

// ---------------------------------------------------------------------------
// Fused GEMM:  out[mat] = A @ W[mat] + b[mat]  for mat in {q,k,v,skip}
// A: [N, IN_C] row-major, W: [IN_C, 128] row-major, out: [N, 128].
// Block = (32,4): 4 waves share one LDS-staged 16-row A tile. Each wave owns
// a 16x32 output slab (two 16x16 WMMA tiles) so every A fragment feeds two
// v_wmma_f32_16x16x4_f32 ops. IN_C is a template constant -> full unroll.
// ---------------------------------------------------------------------------
struct GemmArgs {
  const float* A;
  int N;
  const float* W[4];
  const float* b[4];
  float* out[4];
};

template <int IN_C>
__global__ void k_gemm_wmma(GemmArgs ga) {
  __shared__ float As[16 * (IN_C + 1)];  // +1 stride -> conflict-free reads
  const int m0  = blockIdx.x * 16;
  const int tid = threadIdx.y * 32 + threadIdx.x;

  // Cooperative A-tile stage (16 x IN_C).
#pragma unroll
  for (int i = tid; i < 16 * IN_C; i += 128) {
    int r = i / IN_C, cc = i - r * IN_C;
    int row = m0 + r;
    As[r * (IN_C + 1) + cc] = (row < ga.N) ? ga.A[(size_t)row * IN_C + cc]
                                           : 0.0f;
  }
  __syncthreads();

  const int lane  = threadIdx.x;
  const int lm    = lane & 15;       // row (A) / col (B) within tile
  const int hi    = lane >> 4;       // 0: K={0,1}, 1: K={2,3}
  const int wtile = blockIdx.y * 4 + threadIdx.y;  // 0..15, 32 cols each
  const int mat   = wtile >> 2;
  const int col0  = (wtile & 3) * 32;

  const float* W = ga.W[mat];
  v8f c0 = {0.f, 0.f, 0.f, 0.f, 0.f, 0.f, 0.f, 0.f};
  v8f c1 = {0.f, 0.f, 0.f, 0.f, 0.f, 0.f, 0.f, 0.f};

  const float* bp = W + (size_t)(hi * 2) * 128 + col0 + lm;
  const float* as = &As[lm * (IN_C + 1) + hi * 2];

#pragma unroll
  for (int kk = 0; kk < IN_C; kk += 4) {
    v2f a, b0, b1;
    a.x  = as[0];
    a.y  = as[1];
    b0.x = bp[0];         // B[k][n]
    b0.y = bp[128];       // B[k+1][n]
    b1.x = bp[16];        // B[k][n+16]
    b1.y = bp[128 + 16];  // B[k+1][n+16]
    c0 = __builtin_amdgcn_wmma_f32_16x16x4_f32(false, a, false, b0, (short)0,
                                               c0, false, false);
    c1 = __builtin_amdgcn_wmma_f32_16x16x4_f32(false, a, false, b1, (short)0,
                                               c1, false, false);
    as += 4;
    bp += 4 * 128;
  }

  const float bias0 = ga.b[mat][col0 + lm];
  const float bias1 = ga.b[mat][col0 + 16 + lm];
  float* out = ga.out[mat];
#pragma unroll
  for (int g = 0; g < 8; ++g) {
    int row = m0 + g + hi * 8;  // C layout: vgpr g -> M=g (lanes0-15), M=8+g
    if (row < ga.N) {
      float* op = out + (size_t)row * 128 + col0 + lm;
      op[0]  = c0[g] + bias0;
      op[16] = c1[g] + bias1;
    }
  }
}

// ---------------------------------------------------------------------------
// Per-layer state init: out accumulator = 0, amax = enc(-inf), denom = 0.
// ---------------------------------------------------------------------------
__global__ void k_init_layer(float* outb, unsigned* amax, float* denom,
                             size_t nOut, size_t nHead) {
  size_t i = (size_t)blockIdx.x * blockDim.x + threadIdx.x;
  if (i < nOut) outb[i] = 0.0f;
  if (i < nHead) { amax[i] = ENC_NEG_INF; denom[i] = 0.0f; }
}

__global__ void k_zero(float* p, size_t n) {
  size_t i = (size_t)blockIdx.x * blockDim.x + threadIdx.x;
  if (i < n) p[i] = 0.0f;
}

// ---------------------------------------------------------------------------
// Edge pass A: logits[e,h] = dot(q[dst], k[src]+ea*We) * scale ; segment max.
// One wave per edge; lane handles 4 channels (128 total).
// ---------------------------------------------------------------------------
template <int H>
__global__ void k_edge_logits(const float4* __restrict__ q4,
                              const float4* __restrict__ k4,
                              const int* __restrict__ src,
                              const int* __restrict__ dst,
                              const float* __restrict__ eattr,
                              const float4* __restrict__ We4, float scale,
                              float* __restrict__ logits,
                              unsigned* __restrict__ amax, int E) {
  int e = blockIdx.x * blockDim.y + threadIdx.y;
  if (e >= E) return;
  int lane = threadIdx.x;
  int s = src[e], d = dst[e];
  float ea = eattr[e];
  float4 w  = We4[lane];
  float4 kk = k4[(size_t)s * 32 + lane];
  float4 qq = q4[(size_t)d * 32 + lane];
  float p = qq.x * (kk.x + ea * w.x) + qq.y * (kk.y + ea * w.y) +
            qq.z * (kk.z + ea * w.z) + qq.w * (kk.w + ea * w.w);
  p += __shfl_xor(p, 1, 32);
  p += __shfl_xor(p, 2, 32);
  if (H == 8) {
    if ((lane & 3) == 0) {
      int h = lane >> 2;
      float lg = p * scale;
      logits[(size_t)e * 8 + h] = lg;
      AMAXU(&amax[(size_t)d * 8 + h], encF(lg));
    }
  } else {
    p += __shfl_xor(p, 4, 32);
    p += __shfl_xor(p, 8, 32);
    p += __shfl_xor(p, 16, 32);
    if (lane == 0) {
      float lg = p * scale;
      logits[e] = lg;
      AMAXU(&amax[d], encF(lg));
    }
  }
}

// ---------------------------------------------------------------------------
// Edge pass B: ex = exp(logit - amax[dst]);  denom[dst] += ex  (in-place).
// ---------------------------------------------------------------------------
__global__ void k_edge_exp(float* __restrict__ logits,
                           const unsigned* __restrict__ amax,
                           const int* __restrict__ dst,
                           float* __restrict__ denom, int E, int H) {
  int e = blockIdx.x * blockDim.x + threadIdx.x;
  if (e >= E) return;
  int d = dst[e];
  for (int h = 0; h < H; ++h) {
    float am = decF(amax[(size_t)d * H + h]);
    float ex = expf(logits[(size_t)e * H + h] - am);
    logits[(size_t)e * H + h] = ex;
    AADD(&denom[(size_t)d * H + h], ex);
  }
}

// ---------------------------------------------------------------------------
// Edge pass C: out[dst] += (v[src] + ea*We) * (ex / (denom[dst] + 1e-16)).
// ---------------------------------------------------------------------------
template <int H>
__global__ void k_edge_agg(const float4* __restrict__ v4,
                           const float* __restrict__ ex,
                           const float* __restrict__ denom,
                           const int* __restrict__ src,
                           const int* __restrict__ dst,
                           const float* __restrict__ eattr,
                           const float4* __restrict__ We4,
                           float* __restrict__ outb, int E) {
  int e = blockIdx.x * blockDim.y + threadIdx.y;
  if (e >= E) return;
  int lane = threadIdx.x;
  int s = src[e], d = dst[e];
  float ea = eattr[e];
  int h = (H == 8) ? (lane >> 2) : 0;
  float alpha = ex[(size_t)e * H + h] / (denom[(size_t)d * H + h] + 1e-16f);
  float4 w  = We4[lane];
  float4 vv = v4[(size_t)s * 32 + lane];
  float* op = outb + (size_t)d * 128 + lane * 4;
  AADD(&op[0], (vv.x + ea * w.x) * alpha);
  AADD(&op[1], (vv.y + ea * w.y) * alpha);
  AADD(&op[2], (vv.z + ea * w.z) * alpha);
  AADD(&op[3], (vv.w + ea * w.w) * alpha);
}

// ---------------------------------------------------------------------------
// Beta-gated skip: beta = sigmoid([out, xr, out-xr] @ Wbeta);
// x = beta*xr + (1-beta)*out ; optional ReLU. One wave per node.
// ---------------------------------------------------------------------------
__global__ void k_beta(const float4* __restrict__ out4,
                       const float4* __restrict__ xr4,
                       const float4* __restrict__ Wb4,
                       float4* __restrict__ xnew4, int N, int relu) {
  int n = blockIdx.x * blockDim.y + threadIdx.y;
  if (n >= N) return;
  int lane = threadIdx.x;
  float4 o = out4[(size_t)n * 32 + lane];
  float4 r = xr4[(size_t)n * 32 + lane];
  float4 w1 = Wb4[lane], w2 = Wb4[32 + lane], w3 = Wb4[64 + lane];
  float p = o.x * w1.x + o.y * w1.y + o.z * w1.z + o.w * w1.w +
            r.x * w2.x + r.y * w2.y + r.z * w2.z + r.w * w2.w +
            (o.x - r.x) * w3.x + (o.y - r.y) * w3.y +
            (o.z - r.z) * w3.z + (o.w - r.w) * w3.w;
  p = waveReduceAdd(p);
  float beta = 1.0f / (1.0f + expf(-p));
  float4 x;
  x.x = beta * r.x + (1.0f - beta) * o.x;
  x.y = beta * r.y + (1.0f - beta) * o.y;
  x.z = beta * r.z + (1.0f - beta) * o.z;
  x.w = beta * r.w + (1.0f - beta) * o.w;
  if (relu) {
    x.x = fmaxf(x.x, 0.0f); x.y = fmaxf(x.y, 0.0f);
    x.z = fmaxf(x.z, 0.0f); x.w = fmaxf(x.w, 0.0f);
  }
  xnew4[(size_t)n * 32 + lane] = x;
}

// ---------------------------------------------------------------------------
// LayerNorm over 128 + per-graph mean-pool accumulation. One wave per node.
// ---------------------------------------------------------------------------
__global__ void k_ln_pool(const float4* __restrict__ x4,
                          const int* __restrict__ batch,
                          const float* __restrict__ gamma,
                          const float* __restrict__ lbeta,
                          float* __restrict__ psum, float* __restrict__ pcnt,
                          int N) {
  int n = blockIdx.x * blockDim.y + threadIdx.y;
  if (n >= N) return;
  int lane = threadIdx.x;
  float4 xv = x4[(size_t)n * 32 + lane];
  float mu = waveReduceAdd(xv.x + xv.y + xv.z + xv.w) * (1.0f / 128.0f);
  float dx = xv.x - mu, dy = xv.y - mu, dz = xv.z - mu, dw = xv.w - mu;
  float var = waveReduceAdd(dx * dx + dy * dy + dz * dz + dw * dw) *
              (1.0f / 128.0f);
  float rstd = rsqrtf(var + 1e-5f);
  int b = batch[n];
  int c = lane * 4;
  float* sp = psum + (size_t)b * 128 + c;
  AADD(&sp[0], dx * rstd * gamma[c + 0] + lbeta[c + 0]);
  AADD(&sp[1], dy * rstd * gamma[c + 1] + lbeta[c + 1]);
  AADD(&sp[2], dz * rstd * gamma[c + 2] + lbeta[c + 2]);
  AADD(&sp[3], dw * rstd * gamma[c + 3] + lbeta[c + 3]);
  if (lane == 0) AADD(&pcnt[b], 1.0f);
}

__global__ void k_pool_div(const float* __restrict__ psum,
                           const float* __restrict__ pcnt,
                           float* __restrict__ out) {
  int i = blockIdx.x * blockDim.x + threadIdx.x;
  if (i >= 64 * 128) return;
  float c = pcnt[i >> 7];
  out[i] = (c > 0.0f) ? psum[i] / fmaxf(c, 1.0f) : 0.0f;
}

// ---------------------------------------------------------------------------
// Host orchestration
// ---------------------------------------------------------------------------
extern "C" void kernel_launch(void* const* d_in, const int* in_sizes, int n_in,
                              void* d_out, int out_size, void* d_ws,
                              size_t ws_size, hipStream_t stream) {
  (void)n_in; (void)out_size; (void)ws_size;
  const float* x_in  = (const float*)d_in[0];
  const int*   eidx  = (const int*)d_in[1];
  const float* eattr = (const float*)d_in[2];
  const int*   batch = (const int*)d_in[3];

  const int N = in_sizes[3];       // N_NODES
  const int E = in_sizes[2];       // N_EDGES (edge_dim == 1)
  const int* src = eidx;
  const int* dst = eidx + E;

  // params: 3 layers x 10 leaves; detect pytree flatten order at runtime.
  enum { WQ = 0, BQ, WK, BK, WV, BV, WE, WS, BS, WB };
  const float* P[3][10];
  const bool sortedOrder = (in_sizes[4] == 3 * 128);  // Wbeta first
  int idx = 4;
  for (int l = 0; l < 3; ++l) {
    const float* leaf[10];
    for (int j = 0; j < 10; ++j) leaf[j] = (const float*)d_in[idx + j];
    if (sortedOrder) {  // Wbeta, We, Wk, Wq, Wskip, Wv, bk, bq, bskip, bv
      P[l][WB] = leaf[0]; P[l][WE] = leaf[1]; P[l][WK] = leaf[2];
      P[l][WQ] = leaf[3]; P[l][WS] = leaf[4]; P[l][WV] = leaf[5];
      P[l][BK] = leaf[6]; P[l][BQ] = leaf[7]; P[l][BS] = leaf[8];
      P[l][BV] = leaf[9];
    } else {            // insertion order
      P[l][WQ] = leaf[0]; P[l][BQ] = leaf[1]; P[l][WK] = leaf[2];
      P[l][BK] = leaf[3]; P[l][WV] = leaf[4]; P[l][BV] = leaf[5];
      P[l][WE] = leaf[6]; P[l][WS] = leaf[7]; P[l][BS] = leaf[8];
      P[l][WB] = leaf[9];
    }
    idx += 10;
  }
  const float* ln_g = (const float*)d_in[idx++];
  const float* ln_b = (const float*)d_in[idx++];

  // Workspace carve-up (floats).
  const size_t NF = (size_t)N * 128;
  float* ws     = (float*)d_ws;
  float* q      = ws;
  float* k      = q + NF;
  float* v      = k + NF;
  float* xr     = v + NF;
  float* outb   = xr + NF;
  float* xbuf   = outb + NF;
  float* logits = xbuf + NF;                       // E * 8
  unsigned* amax = (unsigned*)(logits + (size_t)E * 8);   // N * 8
  float* denom  = (float*)(amax + (size_t)N * 8);         // N * 8
  float* psum   = denom + (size_t)N * 8;                  // 64 * 128
  float* pcnt   = psum + 64 * 128;                        // 64

  const int Hs[3]     = {8, 8, 1};
  const float scl[3]  = {0.25f, 0.25f, 0.088388347648318447f};  // 1/sqrt(D)

  const dim3 gemmGrid((N + 15) / 16, 4);
  const dim3 gemmBlk(32, 4);
  const dim3 edgeBlk(32, 8);
  const int  edgeGrid = (E + 7) / 8;
  const dim3 nodeBlk(32, 8);
  const int  nodeGrid = (N + 7) / 8;

  for (int l = 0; l < 3; ++l) {
    const int H = Hs[l];

    // 1) reset per-layer accumulators
    k_init_layer<<<(unsigned)((NF + 255) / 256), 256, 0, stream>>>(
        outb, amax, denom, NF, (size_t)N * 8);

    // 2) fused QKV+skip projections via WMMA fp32
    GemmArgs ga;
    ga.A = (l == 0) ? x_in : xbuf;
    ga.N = N;
    ga.W[0] = P[l][WQ]; ga.W[1] = P[l][WK]; ga.W[2] = P[l][WV]; ga.W[3] = P[l][WS];
    ga.b[0] = P[l][BQ]; ga.b[1] = P[l][BK]; ga.b[2] = P[l][BV]; ga.b[3] = P[l][BS];
    ga.out[0] = q; ga.out[1] = k; ga.out[2] = v; ga.out[3] = xr;
    if (l == 0)
      k_gemm_wmma<64><<<gemmGrid, gemmBlk, 0, stream>>>(ga);
    else
      k_gemm_wmma<128><<<gemmGrid, gemmBlk, 0, stream>>>(ga);

    // 3) edge logits + segment max
    if (H == 8)
      k_edge_logits<8><<<edgeGrid, edgeBlk, 0, stream>>>(
          (const float4*)q, (const float4*)k, src, dst, eattr,
          (const float4*)P[l][WE], scl[l], logits, amax, E);
    else
      k_edge_logits<1><<<edgeGrid, edgeBlk, 0, stream>>>(
          (const float4*)q, (const float4*)k, src, dst, eattr,
          (const float4*)P[l][WE], scl[l], logits, amax, E);

    // 4) exp + segment sum (denominator)
    k_edge_exp<<<(E + 255) / 256, 256, 0, stream>>>(logits, amax, dst, denom,
                                                    E, H);

    // 5) weighted message aggregation
    if (H == 8)
      k_edge_agg<8><<<edgeGrid, edgeBlk, 0, stream>>>(
          (const float4*)v, logits, denom, src, dst, eattr,
          (const float4*)P[l][WE], outb, E);
    else
      k_edge_agg<1><<<edgeGrid, edgeBlk, 0, stream>>>(
          (const float4*)v, logits, denom, src, dst, eattr,
          (const float4*)P[l][WE], outb, E);

    // 6) beta-gated skip (+ ReLU on layers 0,1)
    k_beta<<<nodeGrid, nodeBlk, 0, stream>>>(
        (const float4*)outb, (const float4*)xr, (const float4*)P[l][WB],
        (float4*)xbuf, N, (l < 2) ? 1 : 0);
  }

  // Final: LayerNorm + global mean pool.
  k_zero<<<(64 * 128 + 64 + 255) / 256, 256, 0, stream>>>(
      psum, (size_t)(64 * 128 + 64));
  k_ln_pool<<<nodeGrid, nodeBlk, 0, stream>>>((const float4*)xbuf, batch, ln_g,
                                              ln_b, psum, pcnt, N);
  k_pool_div<<<(64 * 128 + 255) / 256, 256, 0, stream>>>(psum, pcnt,
                                                         (float*)d_out);
}